// AttentionDecoder_13907104104762
// MI455X (gfx1250) — compile-verified
//
#include <hip/hip_runtime.h>
#include <hip/hip_bf16.h>
#include <math.h>

// Problem constants (match reference setup_inputs)
#define BB 64
#define SS 96
#define HH 1024
#define EE 512
#define VV 32000
#define TT 32
#define XH (EE + HH)   // 1536
#define G3 (3 * HH)    // 3072

typedef __attribute__((ext_vector_type(2))) float v2f;
typedef __attribute__((ext_vector_type(8))) float v8f;

#define WMMA_F32(a, b, c) \
    __builtin_amdgcn_wmma_f32_16x16x4_f32(false, (a), false, (b), (short)0, (c), false, false)

// ---------------------------------------------------------------------------
// Generic fp32 WMMA GEMM:  out[64, N] = A[64, K] @ W[N, K]^T  (+ bias | + addC)
// block = 128 threads (4 waves). Each wave owns one 16-col strip of the
// block's 64-column tile and accumulates all four 16-row M tiles (M = 64).
// Inner loop: 4-deep rotating fragment buffers (chunks k-16..k-4 resident,
// each buffer reloaded in place right after use) -> every WMMA waits only on
// loads issued one full body (16 WMMAs) earlier. One branch-free, row-clamped
// global_prefetch_b8 per body. Requires N % 64 == 0 and K % 16 == 0, K >= 32.
// ---------------------------------------------------------------------------
__global__ __launch_bounds__(128)
void gemm64_wmma_f32(const float* __restrict__ A, int lda,
                     const float* __restrict__ W, int ldw,
                     const float* __restrict__ bias,       // may be null
                     const float* __restrict__ addC,       // may be null (overrides bias)
                     int ldadd,
                     float* __restrict__ out, long long ldout,
                     int K)
{
    const int wave  = threadIdx.x >> 5;
    const int lane  = threadIdx.x & 31;
    const int idx   = lane & 15;            // n within tile; also m%16 for A frag
    const int khalf = (lane >> 4) << 1;     // 0 or 2: which K pair this lane holds
    const int hi8   = (lane >> 4) << 3;     // 0 or 8: C/D row-half offset
    const int n     = blockIdx.x * 64 + wave * 16 + idx;

    v8f acc[4];
#pragma unroll
    for (int mt = 0; mt < 4; ++mt) {
        if (addC) {
#pragma unroll
            for (int r = 0; r < 8; ++r)
                acc[mt][r] = addC[(size_t)(mt * 16 + hi8 + r) * (size_t)ldadd + n];
        } else {
            const float binit = bias ? bias[n] : 0.0f;
#pragma unroll
            for (int r = 0; r < 8; ++r) acc[mt][r] = binit;
        }
    }

    const float* wp = W + (size_t)n * (size_t)ldw + khalf;   // B frag: W[n][k+khalf..+1]
    const float* ap = A + (size_t)idx * (size_t)lda + khalf; // A frag row base
    const size_t as = (size_t)16 * (size_t)lda;              // A row-tile step

    // ---- prologue: load chunks 0,4,8,12 into the four rotating buffer sets
    v2f b0 = *(const v2f*)(wp + 0);
    v2f a00 = *(const v2f*)(ap + 0),  a01 = *(const v2f*)(ap + as + 0);
    v2f a02 = *(const v2f*)(ap + 2 * as + 0), a03 = *(const v2f*)(ap + 3 * as + 0);
    v2f b1 = *(const v2f*)(wp + 4);
    v2f a10 = *(const v2f*)(ap + 4),  a11 = *(const v2f*)(ap + as + 4);
    v2f a12 = *(const v2f*)(ap + 2 * as + 4), a13 = *(const v2f*)(ap + 3 * as + 4);
    v2f b2 = *(const v2f*)(wp + 8);
    v2f a20 = *(const v2f*)(ap + 8),  a21 = *(const v2f*)(ap + as + 8);
    v2f a22 = *(const v2f*)(ap + 2 * as + 8), a23 = *(const v2f*)(ap + 3 * as + 8);
    v2f b3 = *(const v2f*)(wp + 12);
    v2f a30 = *(const v2f*)(ap + 12), a31 = *(const v2f*)(ap + as + 12);
    v2f a32 = *(const v2f*)(ap + 2 * as + 12), a33 = *(const v2f*)(ap + 3 * as + 12);

    for (int k = 16; k < K; k += 16) {
        // branch-free, row-clamped stream prefetch (~700B ahead in this lane's W row)
        int pk = k + 176; pk = (pk < K - 4) ? pk : (K - 4);
        __builtin_prefetch(wp + pk, 0, 3);

        // group 0: compute chunk k-16, reload buffer with chunk k
        acc[0] = WMMA_F32(a00, b0, acc[0]);
        acc[1] = WMMA_F32(a01, b0, acc[1]);
        acc[2] = WMMA_F32(a02, b0, acc[2]);
        acc[3] = WMMA_F32(a03, b0, acc[3]);
        b0 = *(const v2f*)(wp + k);
        a00 = *(const v2f*)(ap + k);          a01 = *(const v2f*)(ap + as + k);
        a02 = *(const v2f*)(ap + 2 * as + k); a03 = *(const v2f*)(ap + 3 * as + k);

        // group 1: compute chunk k-12, reload with chunk k+4
        acc[0] = WMMA_F32(a10, b1, acc[0]);
        acc[1] = WMMA_F32(a11, b1, acc[1]);
        acc[2] = WMMA_F32(a12, b1, acc[2]);
        acc[3] = WMMA_F32(a13, b1, acc[3]);
        b1 = *(const v2f*)(wp + k + 4);
        a10 = *(const v2f*)(ap + k + 4);          a11 = *(const v2f*)(ap + as + k + 4);
        a12 = *(const v2f*)(ap + 2 * as + k + 4); a13 = *(const v2f*)(ap + 3 * as + k + 4);

        // group 2: compute chunk k-8, reload with chunk k+8
        acc[0] = WMMA_F32(a20, b2, acc[0]);
        acc[1] = WMMA_F32(a21, b2, acc[1]);
        acc[2] = WMMA_F32(a22, b2, acc[2]);
        acc[3] = WMMA_F32(a23, b2, acc[3]);
        b2 = *(const v2f*)(wp + k + 8);
        a20 = *(const v2f*)(ap + k + 8);          a21 = *(const v2f*)(ap + as + k + 8);
        a22 = *(const v2f*)(ap + 2 * as + k + 8); a23 = *(const v2f*)(ap + 3 * as + k + 8);

        // group 3: compute chunk k-4, reload with chunk k+12
        acc[0] = WMMA_F32(a30, b3, acc[0]);
        acc[1] = WMMA_F32(a31, b3, acc[1]);
        acc[2] = WMMA_F32(a32, b3, acc[2]);
        acc[3] = WMMA_F32(a33, b3, acc[3]);
        b3 = *(const v2f*)(wp + k + 12);
        a30 = *(const v2f*)(ap + k + 12);          a31 = *(const v2f*)(ap + as + k + 12);
        a32 = *(const v2f*)(ap + 2 * as + k + 12); a33 = *(const v2f*)(ap + 3 * as + k + 12);
    }

    // ---- epilogue: compute the last four resident chunks
    acc[0] = WMMA_F32(a00, b0, acc[0]);
    acc[1] = WMMA_F32(a01, b0, acc[1]);
    acc[2] = WMMA_F32(a02, b0, acc[2]);
    acc[3] = WMMA_F32(a03, b0, acc[3]);
    acc[0] = WMMA_F32(a10, b1, acc[0]);
    acc[1] = WMMA_F32(a11, b1, acc[1]);
    acc[2] = WMMA_F32(a12, b1, acc[2]);
    acc[3] = WMMA_F32(a13, b1, acc[3]);
    acc[0] = WMMA_F32(a20, b2, acc[0]);
    acc[1] = WMMA_F32(a21, b2, acc[1]);
    acc[2] = WMMA_F32(a22, b2, acc[2]);
    acc[3] = WMMA_F32(a23, b2, acc[3]);
    acc[0] = WMMA_F32(a30, b3, acc[0]);
    acc[1] = WMMA_F32(a31, b3, acc[1]);
    acc[2] = WMMA_F32(a32, b3, acc[2]);
    acc[3] = WMMA_F32(a33, b3, acc[3]);

#pragma unroll
    for (int mt = 0; mt < 4; ++mt)
#pragma unroll
        for (int r = 0; r < 8; ++r)
            out[(size_t)(mt * 16 + hi8 + r) * (size_t)ldout + n] = acc[mt][r];
}

// ---------------------------------------------------------------------------
// u[h] = sum_g Va[g] * Ua[g][h]   (Ua^T @ Va) -- collapses the keys@Ua^T GEMM
// ---------------------------------------------------------------------------
__global__ void ua_va_kernel(const float* __restrict__ Ua,
                             const float* __restrict__ Va,
                             float* __restrict__ u)
{
    const int h = blockIdx.x * blockDim.x + threadIdx.x;
    if (h >= HH) return;
    float acc = 0.0f;
    for (int g = 0; g < HH; ++g) acc += Va[g] * Ua[(size_t)g * HH + h];
    u[h] = acc;
}

// ---------------------------------------------------------------------------
// Per batch row b: scores[s] = keys[b,s,:].u ; softmax ; ctx[b,:] = w @ keys[b]
// (attention weights are step-invariant: softmax is shift-invariant w.r.t. the
//  Va.q term, so this runs ONCE, not per step.)
// ---------------------------------------------------------------------------
__global__ __launch_bounds__(256)
void attn_kernel(const float* __restrict__ keys, const float* __restrict__ u,
                 float* __restrict__ wAttn, float* __restrict__ ctx)
{
    __shared__ float sw[SS];
    const int b = blockIdx.x, tid = threadIdx.x;
    const int wave = tid >> 5, lane = tid & 31;
    const float* kb = keys + (size_t)b * SS * HH;

    for (int s = wave; s < SS; s += 8) {
        const float* row = kb + (size_t)s * HH;
        float acc = 0.0f;
        for (int h = lane; h < HH; h += 32) acc += row[h] * u[h];
        for (int off = 16; off > 0; off >>= 1) acc += __shfl_down(acc, off, 32);
        if (lane == 0) sw[s] = acc;
    }
    __syncthreads();
    if (tid == 0) {
        float mx = -INFINITY;
        for (int s = 0; s < SS; ++s) mx = fmaxf(mx, sw[s]);
        float sum = 0.0f;
        for (int s = 0; s < SS; ++s) { float e = expf(sw[s] - mx); sw[s] = e; sum += e; }
        const float inv = 1.0f / sum;
        for (int s = 0; s < SS; ++s) sw[s] *= inv;
    }
    __syncthreads();
    if (tid < SS) wAttn[b * SS + tid] = sw[tid];
    for (int h = tid; h < HH; h += 256) {
        float acc = 0.0f;
        for (int s = 0; s < SS; ++s) acc += sw[s] * kb[(size_t)s * HH + h];
        ctx[(size_t)b * HH + h] = acc;
    }
}

// attn_out[b,t,s] = wAttn[b,s] for all t
__global__ void attn_repl_kernel(const float* __restrict__ wAttn,
                                 float* __restrict__ attn_out)
{
    const int i = blockIdx.x * blockDim.x + threadIdx.x;
    if (i >= BB * TT * SS) return;
    const int s = i % SS;
    const int b = i / (SS * TT);
    attn_out[i] = wAttn[b * SS + s];
}

// ---------------------------------------------------------------------------
// tok[b] = (first ? START : argmax_v logits_prev[b,:]) ; emb[b,:] = relu(table[tok])
// jnp.argmax tie rule: first (lowest) index wins.
// ---------------------------------------------------------------------------
__global__ __launch_bounds__(256)
void argmax_embed_kernel(int first,
                         const float* __restrict__ logits_prev, long long lrow,
                         const float* __restrict__ emb_table,
                         float* __restrict__ emb, int* __restrict__ tok)
{
    __shared__ float sval[256];
    __shared__ int   sidx[256];
    __shared__ int   stok;
    const int b = blockIdx.x, tid = threadIdx.x;

    if (first) {
        if (tid == 0) { stok = 1; tok[b] = 1; }   // START_TOKEN = 1
    } else {
        const float* lr = logits_prev + (size_t)b * (size_t)lrow;
        float bv = -INFINITY; int bi = 0;
        for (int v = tid; v < VV; v += 256) {
            const float x = lr[v];
            if (x > bv) { bv = x; bi = v; }       // strided scan: ties keep lower idx
        }
        sval[tid] = bv; sidx[tid] = bi;
        __syncthreads();
        for (int off = 128; off > 0; off >>= 1) {
            if (tid < off) {
                const float ov = sval[tid + off]; const int oi = sidx[tid + off];
                if (ov > sval[tid] || (ov == sval[tid] && oi < sidx[tid])) {
                    sval[tid] = ov; sidx[tid] = oi;
                }
            }
            __syncthreads();
        }
        if (tid == 0) { stok = sidx[0]; tok[b] = sidx[0]; }
    }
    __syncthreads();
    const float* er = emb_table + (size_t)stok * EE;
    for (int e = tid; e < EE; e += 256)
        emb[(size_t)b * EE + e] = fmaxf(er[e], 0.0f);
}

// ---------------------------------------------------------------------------
// GRU gates (torch order r,z,n): gi/gh already include their biases.
// ---------------------------------------------------------------------------
__global__ __launch_bounds__(256)
void gru_gates_kernel(const float* __restrict__ gi, const float* __restrict__ gh,
                      const float* __restrict__ h, float* __restrict__ hnew)
{
    const int i = blockIdx.x * blockDim.x + threadIdx.x;
    if (i >= BB * HH) return;
    const int b = i / HH, uu = i % HH;
    const float* gib = gi + (size_t)b * G3;
    const float* ghb = gh + (size_t)b * G3;
    const float r = 1.0f / (1.0f + expf(-(gib[uu] + ghb[uu])));
    const float z = 1.0f / (1.0f + expf(-(gib[HH + uu] + ghb[HH + uu])));
    const float n = tanhf(gib[2 * HH + uu] + r * ghb[2 * HH + uu]);
    hnew[i] = (1.0f - z) * n + z * h[i];
}

// ---------------------------------------------------------------------------
extern "C" void kernel_launch(void* const* d_in, const int* in_sizes, int n_in,
                              void* d_out, int out_size, void* d_ws, size_t ws_size,
                              hipStream_t stream)
{
    const float* keys      = (const float*)d_in[0];   // [B,S,H]
    const float* enc_h     = (const float*)d_in[1];   // [1,B,H]
    const float* emb_table = (const float*)d_in[2];   // [V,E]
    // d_in[3] = Wa: mathematically dead (softmax shift invariance kills the Va.q term)
    const float* Ua        = (const float*)d_in[4];   // [H,H]
    const float* Va        = (const float*)d_in[5];   // [H]
    const float* W_ih      = (const float*)d_in[6];   // [3H, E+H]
    const float* W_hh      = (const float*)d_in[7];   // [3H, H]
    const float* b_ih      = (const float*)d_in[8];   // [3H]
    const float* b_hh      = (const float*)d_in[9];   // [3H]
    const float* W_out     = (const float*)d_in[10];  // [V,H]
    const float* b_out     = (const float*)d_in[11];  // [V]
    (void)in_sizes; (void)n_in; (void)out_size; (void)ws_size;

    float* out        = (float*)d_out;
    float* logits_out = out;                                   // [B,T,V]
    float* hT_out     = out + (size_t)BB * TT * VV;            // [1,B,H]
    float* attn_out   = hT_out + (size_t)BB * HH;              // [B,T,S]

    // Workspace carve-up (~3.3 MB)
    float* wsp    = (float*)d_ws;
    float* u      = wsp;                 // 1024
    float* wAttn  = u + HH;              // B*S
    float* ctx    = wAttn + BB * SS;     // B*H
    float* gi_ctx = ctx + BB * HH;       // B*3H  (ctx part of gi + b_ih, step-invariant)
    float* emb    = gi_ctx + BB * G3;    // B*E
    float* gi     = emb + BB * EE;       // B*3H
    float* gh     = gi + BB * G3;        // B*3H
    float* h0     = gh + BB * G3;        // B*H
    float* h1     = h0 + BB * HH;        // B*H
    int*   tok    = (int*)(h1 + BB * HH);// B

    // ---- one-time precompute ----
    ua_va_kernel<<<(HH + 255) / 256, 256, 0, stream>>>(Ua, Va, u);
    attn_kernel<<<BB, 256, 0, stream>>>(keys, u, wAttn, ctx);
    attn_repl_kernel<<<(BB * TT * SS + 255) / 256, 256, 0, stream>>>(wAttn, attn_out);
    // gi_ctx = ctx @ W_ih[:, E:]^T + b_ih
    gemm64_wmma_f32<<<G3 / 64, 128, 0, stream>>>(ctx, HH, W_ih + EE, XH,
                                                 b_ih, nullptr, 0,
                                                 gi_ctx, (long long)G3, HH);
    hipMemcpyAsync(h0, enc_h, (size_t)BB * HH * sizeof(float),
                   hipMemcpyDeviceToDevice, stream);

    float* hcur = h0;
    float* hnext = h1;

    for (int t = 0; t < TT; ++t) {
        const float* prev = logits_out + (size_t)(t > 0 ? t - 1 : 0) * VV;
        argmax_embed_kernel<<<BB, 256, 0, stream>>>(t == 0 ? 1 : 0, prev,
                                                    (long long)TT * VV,
                                                    emb_table, emb, tok);
        // gi = emb @ W_ih[:, :E]^T + gi_ctx
        gemm64_wmma_f32<<<G3 / 64, 128, 0, stream>>>(emb, EE, W_ih, XH,
                                                     nullptr, gi_ctx, G3,
                                                     gi, (long long)G3, EE);
        // gh = h @ W_hh^T + b_hh
        gemm64_wmma_f32<<<G3 / 64, 128, 0, stream>>>(hcur, HH, W_hh, HH,
                                                     b_hh, nullptr, 0,
                                                     gh, (long long)G3, HH);
        gru_gates_kernel<<<(BB * HH + 255) / 256, 256, 0, stream>>>(gi, gh, hcur, hnext);
        // logits[:, t, :] = h_new @ W_out^T + b_out
        gemm64_wmma_f32<<<VV / 64, 128, 0, stream>>>(hnext, HH, W_out, HH,
                                                     b_out, nullptr, 0,
                                                     logits_out + (size_t)t * VV,
                                                     (long long)TT * VV, HH);
        float* tmp = hcur; hcur = hnext; hnext = tmp;
    }

    hipMemcpyAsync(hT_out, hcur, (size_t)BB * HH * sizeof(float),
                   hipMemcpyDeviceToDevice, stream);
}